// DSVDD_44083544326457
// MI455X (gfx1250) — compile-verified
//
#include <hip/hip_runtime.h>
#include <math.h>

typedef float v2f __attribute__((ext_vector_type(2)));
typedef float v8f __attribute__((ext_vector_type(8)));

#define DIM   1792
#define HW    4096
#define NCENT 4096
#define NB    4
#define WROW  1794   // DIM + 2 coord channels
#define ALPHA_C 0.1f
// loss scale: 1 / (NU * B * HW * 3)  with NU = 0.001
#define LOSS_SCALE (1.0f / (0.001f * (float)NB * (float)HW * 3.0f))

// ---------------------------------------------------------------------------
// Kernel 1: phiT[b, o, hw] = sum_i W[o,i] * sample[b,i,hw]  (+ coords + bias)
// GEMM with M = o (so stores are coalesced), N = hw, K = feature channel.
// Block = 256 threads (8 waves), tile 128(o) x 128(hw), wave = 2x4 subtiles.
// ---------------------------------------------------------------------------
__global__ __launch_bounds__(256)
void phi_gemm_kernel(const float* __restrict__ sample,
                     const float* __restrict__ W,
                     const float* __restrict__ bias,
                     float* __restrict__ phiT)
{
    const int lane  = threadIdx.x & 31;
    const int wv    = threadIdx.x >> 5;
    const int wm    = wv & 3;        // row group (o), 32 rows each
    const int wn    = wv >> 2;       // col group (hw), 64 cols each
    const int o0    = blockIdx.x * 128;
    const int hw0   = blockIdx.y * 128;
    const int bb    = blockIdx.z;
    const int lhalf = lane >> 4;     // 0: K,K+1  1: K+2,K+3
    const int lmod  = lane & 15;

    v8f acc[2][4];
    #pragma unroll
    for (int tm = 0; tm < 2; ++tm)
        #pragma unroll
        for (int tn = 0; tn < 4; ++tn)
            acc[tm][tn] = (v8f){0.f,0.f,0.f,0.f,0.f,0.f,0.f,0.f};

    const float* samp_b = sample + (size_t)bb * DIM * HW;

    for (int k = 0; k < DIM; k += 4) {
        const int kk = k + 2 * lhalf;
        v2f afrag[2];
        #pragma unroll
        for (int tm = 0; tm < 2; ++tm) {
            int row = o0 + wm * 32 + tm * 16 + lmod;        // o (M)
            afrag[tm] = *(const v2f*)(W + (size_t)row * WROW + kk); // 8B aligned
        }
        v2f bfrag[4];
        #pragma unroll
        for (int tn = 0; tn < 4; ++tn) {
            int col = hw0 + wn * 64 + tn * 16 + lmod;       // hw (N)
            bfrag[tn].x = samp_b[(size_t)kk       * HW + col];
            bfrag[tn].y = samp_b[(size_t)(kk + 1) * HW + col];
        }
        #pragma unroll
        for (int tm = 0; tm < 2; ++tm)
            #pragma unroll
            for (int tn = 0; tn < 4; ++tn)
                acc[tm][tn] = __builtin_amdgcn_wmma_f32_16x16x4_f32(
                    false, afrag[tm], false, bfrag[tn],
                    (short)0, acc[tm][tn], false, false);
    }

    // Epilogue: bias + coord channels (xx,yy are linspace(-1,1,64) -> step 2/63)
    #pragma unroll
    for (int tm = 0; tm < 2; ++tm) {
        #pragma unroll
        for (int tn = 0; tn < 4; ++tn) {
            int col = hw0 + wn * 64 + tn * 16 + lmod;   // hw
            float xx = -1.0f + (2.0f / 63.0f) * (float)(col & 63);
            float yy = -1.0f + (2.0f / 63.0f) * (float)(col >> 6);
            #pragma unroll
            for (int r = 0; r < 8; ++r) {
                int o = o0 + wm * 32 + tm * 16 + r + 8 * lhalf;  // M = r (+8 hi half)
                float v = acc[tm][tn][r] + bias[o]
                        + xx * W[(size_t)o * WROW + DIM]
                        + yy * W[(size_t)o * WROW + DIM + 1];
                phiT[((size_t)bb * DIM + o) * HW + col] = v;
            }
        }
    }
}

// ---------------------------------------------------------------------------
// Kernel 2: feat2[b*HW + hw] = sum_o phiT[b,o,hw]^2   (coalesced across hw)
// ---------------------------------------------------------------------------
__global__ void feat2_kernel(const float* __restrict__ phiT,
                             float* __restrict__ feat2)
{
    int idx = blockIdx.x * blockDim.x + threadIdx.x;   // [0, NB*HW)
    int bb = idx >> 12;
    int hw = idx & (HW - 1);
    const float* p = phiT + (size_t)bb * DIM * HW + hw;
    float s = 0.f;
    for (int o = 0; o < DIM; ++o) {
        float v = p[(size_t)o * HW];
        s += v * v;
    }
    feat2[idx] = s;
}

// ---------------------------------------------------------------------------
// Kernel 3: cent2[n] = sum_k C[k,n]^2 ; also zero the loss accumulator out[0]
// ---------------------------------------------------------------------------
__global__ void cent2_kernel(const float* __restrict__ C,
                             float* __restrict__ cent2,
                             float* __restrict__ out)
{
    int n = blockIdx.x * blockDim.x + threadIdx.x;
    float s = 0.f;
    for (int k = 0; k < DIM; ++k) {
        float v = C[(size_t)k * NCENT + n];
        s += v * v;
    }
    cent2[n] = s;
    if (n == 0) out[0] = 0.0f;
}

// ---------------------------------------------------------------------------
// Kernel 4: fused dist2 GEMM + per-row top-6 + score + loss.
// Block = 256 threads (8 waves), owns 128 rows (hw) of one batch; loops over
// all 4096 centers in 64-wide chunks. Wave = 2(row)x2(col) 16x16 WMMA tiles.
// dist2 tile staged in LDS; threads 0..127 keep per-row sorted top-6 in regs.
// ---------------------------------------------------------------------------
__global__ __launch_bounds__(256)
void dist_topk_kernel(const float* __restrict__ phiT,
                      const float* __restrict__ C,
                      const float* __restrict__ feat2,
                      const float* __restrict__ cent2,
                      const float* __restrict__ rptr,
                      float* __restrict__ out)   // out[0]=loss, out[1..]=score
{
    __shared__ float tile[128 * 64];
    __shared__ float red[128];

    const int lane  = threadIdx.x & 31;
    const int wv    = threadIdx.x >> 5;
    const int wm    = wv & 3;      // row group of 32 (hw)
    const int wn    = wv >> 2;     // col group of 32 (centers)
    const int hw0   = blockIdx.x * 128;
    const int bb    = blockIdx.y;
    const int lhalf = lane >> 4;
    const int lmod  = lane & 15;

    const float* phi_b = phiT + (size_t)bb * DIM * HW;

    float topk[6];
    #pragma unroll
    for (int i = 0; i < 6; ++i) topk[i] = 3.0e38f;

    for (int n0 = 0; n0 < NCENT; n0 += 64) {
        v8f acc[2][2];
        #pragma unroll
        for (int tm = 0; tm < 2; ++tm)
            #pragma unroll
            for (int tn = 0; tn < 2; ++tn)
                acc[tm][tn] = (v8f){0.f,0.f,0.f,0.f,0.f,0.f,0.f,0.f};

        for (int k = 0; k < DIM; k += 4) {
            const int kk = k + 2 * lhalf;
            v2f afrag[2];
            #pragma unroll
            for (int tm = 0; tm < 2; ++tm) {
                int row = hw0 + wm * 32 + tm * 16 + lmod;   // hw (M)
                afrag[tm].x = phi_b[(size_t)kk       * HW + row];
                afrag[tm].y = phi_b[(size_t)(kk + 1) * HW + row];
            }
            v2f bfrag[2];
            #pragma unroll
            for (int tn = 0; tn < 2; ++tn) {
                int col = n0 + wn * 32 + tn * 16 + lmod;    // center (N)
                bfrag[tn].x = C[(size_t)kk       * NCENT + col];
                bfrag[tn].y = C[(size_t)(kk + 1) * NCENT + col];
            }
            #pragma unroll
            for (int tm = 0; tm < 2; ++tm)
                #pragma unroll
                for (int tn = 0; tn < 2; ++tn)
                    acc[tm][tn] = __builtin_amdgcn_wmma_f32_16x16x4_f32(
                        false, afrag[tm], false, bfrag[tn],
                        (short)0, acc[tm][tn], false, false);
        }

        // dist2 = |phi|^2 + |C|^2 - 2*dot  -> LDS tile (128 rows x 64 cols)
        #pragma unroll
        for (int tm = 0; tm < 2; ++tm) {
            #pragma unroll
            for (int tn = 0; tn < 2; ++tn) {
                int coln = wn * 32 + tn * 16 + lmod;        // local col
                float c2 = cent2[n0 + coln];
                #pragma unroll
                for (int r = 0; r < 8; ++r) {
                    int rowl = wm * 32 + tm * 16 + r + 8 * lhalf;
                    float f2 = feat2[bb * HW + hw0 + rowl];
                    tile[rowl * 64 + coln] = f2 + c2 - 2.0f * acc[tm][tn][r];
                }
            }
        }
        __syncthreads();

        // per-row top-6 (ascending) maintained in registers across chunks
        if (threadIdx.x < 128) {
            const float* rowp = &tile[threadIdx.x * 64];
            for (int j = 0; j < 64; ++j) {
                float v = rowp[j];
                if (v < topk[5]) {
                    topk[5] = v;
                    #pragma unroll
                    for (int t = 5; t > 0; --t) {
                        if (topk[t] < topk[t - 1]) {
                            float tmp = topk[t];
                            topk[t] = topk[t - 1];
                            topk[t - 1] = tmp;
                        }
                    }
                }
            }
        }
        __syncthreads();
    }

    // score (softmin over 3 smallest sqrt-distances) + loss partial
    float part = 0.f;
    if (threadIdx.x < 128) {
        int grow = bb * HW + hw0 + (int)threadIdx.x;
        float d0 = sqrtf(fmaxf(topk[0], 0.f));
        float d1 = sqrtf(fmaxf(topk[1], 0.f));
        float d2 = sqrtf(fmaxf(topk[2], 0.f));
        // softmax(-d)[0] with max-shift at -d0 (d0 is the smallest)
        float w0 = 1.0f / (1.0f + expf(d0 - d1) + expf(d0 - d2));
        out[1 + grow] = w0 * d0;

        float rr = rptr[0];
        float r2 = rr * rr;
        part = fmaxf(topk[0] - r2, 0.f) + fmaxf(topk[1] - r2, 0.f)
             + fmaxf(topk[2] - r2, 0.f)
             + fmaxf(r2 - topk[3] - ALPHA_C, 0.f)
             + fmaxf(r2 - topk[4] - ALPHA_C, 0.f)
             + fmaxf(r2 - topk[5] - ALPHA_C, 0.f);
        red[threadIdx.x] = part;
    }
    __syncthreads();
    if (threadIdx.x == 0) {
        float t = 0.f;
        for (int i = 0; i < 128; ++i) t += red[i];
        atomicAdd(out, t * LOSS_SCALE);
    }
}

// ---------------------------------------------------------------------------
extern "C" void kernel_launch(void* const* d_in, const int* in_sizes, int n_in,
                              void* d_out, int out_size, void* d_ws, size_t ws_size,
                              hipStream_t stream)
{
    (void)in_sizes; (void)n_in; (void)out_size; (void)ws_size;
    const float* sample = (const float*)d_in[0];   // [4,1792,64,64]
    const float* W      = (const float*)d_in[1];   // [1792,1794]
    const float* bias   = (const float*)d_in[2];   // [1792]
    const float* C      = (const float*)d_in[3];   // [1792,4096]
    const float* r      = (const float*)d_in[4];   // [1]
    float* out = (float*)d_out;                    // [1 + 4*4096]

    float* phiT  = (float*)d_ws;                         // NB*DIM*HW floats
    float* feat2 = phiT + (size_t)NB * DIM * HW;         // NB*HW floats
    float* cent2 = feat2 + (size_t)NB * HW;              // NCENT floats

    phi_gemm_kernel<<<dim3(DIM / 128, HW / 128, NB), 256, 0, stream>>>(
        sample, W, bias, phiT);
    feat2_kernel<<<dim3(NB * HW / 256), 256, 0, stream>>>(phiT, feat2);
    cent2_kernel<<<dim3(NCENT / 256), 256, 0, stream>>>(C, cent2, out);
    dist_topk_kernel<<<dim3(HW / 128, NB), 256, 0, stream>>>(
        phiT, C, feat2, cent2, r, out);
}